// ERIC_21105469293013
// MI455X (gfx1250) — compile-verified
//
#include <hip/hip_runtime.h>
#include <hip/hip_bf16.h>
#include <math.h>

// ---------------------------------------------------------------------------
// ERIC forward on MI455X (gfx1250, wave32).
//
// Roofline: edge scatter dominates (~4.9 GB HBM traffic ~= 210us @ 23.3 TB/s).
// All node GEMMs (~5 GFLOP total) run on V_WMMA_F32_16X16X4_F32: one wave owns
// a 16-node x 64-feature tile, K=64 as 16 wmma k-steps, N=64 as 4 n-tiles.
// ---------------------------------------------------------------------------

#define N_NODES 50000
#define N_EDGES 1600000
#define TILES_M 3125          // 50000 / 16 (exact)
#define GEMM_BLOCKS 391       // ceil(3125 / 8 waves)
#define XPAD 68               // padded LDS row stride (floats)

typedef float v2f __attribute__((ext_vector_type(2)));
typedef float v8f __attribute__((ext_vector_type(8)));

__device__ __forceinline__ v8f wmma_f32_16x16x4(v2f a, v2f b, v8f c) {
  // 8 args: (neg_a, A, neg_b, B, c_mod, C, reuse_a, reuse_b)
  return __builtin_amdgcn_wmma_f32_16x16x4_f32(false, a, false, b,
                                               (short)0, c, false, false);
}

// -------------------------------- utility ---------------------------------

__global__ void eric_zero_kernel(float* __restrict__ p, int n) {
  int i = blockIdx.x * 256 + threadIdx.x;
  if (i < n) p[i] = 0.f;
}

// --------------------------- edge scatter (GIN agg) ------------------------
// agg[dst] += x[src]; one thread = one edge x 4 features (float4 gather).
__global__ __launch_bounds__(256) void eric_scatter_kernel(
    const float* __restrict__ X, int xstride,
    const int* __restrict__ E, float* __restrict__ AGG) {
  int idx = blockIdx.x * 256 + threadIdx.x;       // 25.6M threads
  if (idx >= N_EDGES * 16) return;
  int e = idx >> 4;
  int p = idx & 15;
  int src = E[e];
  int dst = E[N_EDGES + e];
  const float4 v =
      *reinterpret_cast<const float4*>(X + (size_t)src * xstride + p * 4);
  float* a = AGG + (size_t)dst * 64 + p * 4;
  atomicAdd(a + 0, v.x);
  atomicAdd(a + 1, v.y);
  atomicAdd(a + 2, v.z);
  atomicAdd(a + 3, v.w);
}

// ------------------------- fused GIN MLP (WMMA f32) ------------------------
// H = relu((2*X + AGG) @ W1 + b1) @ W2 + b2 ; optional BN stat accumulation.
__global__ __launch_bounds__(256) void eric_gin_gemm_kernel(
    const float* __restrict__ X, int xstride, const float* __restrict__ AGG,
    const float* __restrict__ W1, const float* __restrict__ B1,
    const float* __restrict__ W2, const float* __restrict__ B2,
    float* __restrict__ Hout, float* __restrict__ stats) {
  __shared__ float sW[64 * 64];
  __shared__ float sB1[64], sB2[64];
  __shared__ float sX[8][16 * XPAD];   // per-wave 16x64 tile (padded)
  __shared__ float sStat[128];

  const int tid = threadIdx.x, wid = tid >> 5, lane = tid & 31;
  const int half = lane >> 4, l15 = lane & 15;

  if (stats && tid < 128) sStat[tid] = 0.f;
  for (int i = tid; i < 4096; i += 256) sW[i] = W1[i];
  if (tid < 64) { sB1[tid] = B1[tid]; sB2[tid] = B2[tid]; }
  __syncthreads();

  const int tile = blockIdx.x * 8 + wid;
  const bool active = tile < TILES_M;   // wave-uniform: EXEC all-1 for WMMA
  float* sx = &sX[wid][0];

  if (active) {
    const int node0 = tile * 16;
    if (tile + 8 < TILES_M)  // prefetch next block's tile (global_prefetch_b8)
      __builtin_prefetch(X + (size_t)((tile + 8) * 16 + l15) * xstride + half * 32, 0, 0);
    // stage xin = 2*x + agg  (16 rows x 16 float4, coalesced)
    for (int i = lane; i < 256; i += 32) {
      int r = i >> 4, c4 = i & 15;
      const float4 xv = *reinterpret_cast<const float4*>(
          X + (size_t)(node0 + r) * xstride + c4 * 4);
      const float4 av = *reinterpret_cast<const float4*>(
          AGG + (size_t)(node0 + r) * 64 + c4 * 4);
      sx[r * XPAD + c4 * 4 + 0] = 2.f * xv.x + av.x;
      sx[r * XPAD + c4 * 4 + 1] = 2.f * xv.y + av.y;
      sx[r * XPAD + c4 * 4 + 2] = 2.f * xv.z + av.z;
      sx[r * XPAD + c4 * 4 + 3] = 2.f * xv.w + av.w;
    }
    // GEMM1: 16x64 @ 64x64 via 16 k-steps x 4 n-tiles of wmma f32 16x16x4
    v8f acc[4] = {{}, {}, {}, {}};
#pragma unroll
    for (int kk = 0; kk < 16; ++kk) {
      const int kb = kk * 4 + half * 2;
      v2f a; a.x = sx[l15 * XPAD + kb]; a.y = sx[l15 * XPAD + kb + 1];
#pragma unroll
      for (int nt = 0; nt < 4; ++nt) {
        const int col = nt * 16 + l15;
        v2f b; b.x = sW[kb * 64 + col]; b.y = sW[(kb + 1) * 64 + col];
        acc[nt] = wmma_f32_16x16x4(a, b, acc[nt]);
      }
    }
    // relu(acc + b1) back into the same LDS tile (C/D layout: VGPR i ->
    // rows i / i+8 split over lane halves)
#pragma unroll
    for (int nt = 0; nt < 4; ++nt)
#pragma unroll
      for (int i = 0; i < 8; ++i) {
        const int m = i + half * 8;
        const int n = nt * 16 + l15;
        sx[m * XPAD + n] = fmaxf(acc[nt][i] + sB1[n], 0.f);
      }
  }
  __syncthreads();
  for (int i = tid; i < 4096; i += 256) sW[i] = W2[i];  // swap weights
  __syncthreads();
  if (active) {
    const int node0 = tile * 16;
    v8f acc[4] = {{}, {}, {}, {}};
#pragma unroll
    for (int kk = 0; kk < 16; ++kk) {
      const int kb = kk * 4 + half * 2;
      v2f a; a.x = sx[l15 * XPAD + kb]; a.y = sx[l15 * XPAD + kb + 1];
#pragma unroll
      for (int nt = 0; nt < 4; ++nt) {
        const int col = nt * 16 + l15;
        v2f b; b.x = sW[kb * 64 + col]; b.y = sW[(kb + 1) * 64 + col];
        acc[nt] = wmma_f32_16x16x4(a, b, acc[nt]);
      }
    }
#pragma unroll
    for (int nt = 0; nt < 4; ++nt)
#pragma unroll
      for (int i = 0; i < 8; ++i) {
        const int m = i + half * 8;
        const int n = nt * 16 + l15;
        const float h = acc[nt][i] + sB2[n];
        Hout[(size_t)(node0 + m) * 64 + n] = h;
        if (stats) {
          atomicAdd(&sStat[n], h);
          atomicAdd(&sStat[64 + n], h * h);
        }
      }
  }
  if (stats) {
    __syncthreads();
    if (tid < 128) atomicAdd(&stats[tid], sStat[tid]);
  }
}

// ------------------------- BN / LN + affine + relu -------------------------
// BN (bnstats != null): mu/var from accumulated sums over all nodes.
// LN: per-node over 64 features (64-thread groups, tree reduce).
__global__ __launch_bounds__(256) void eric_norm_kernel(
    const float* __restrict__ H, const float* __restrict__ gg,
    const float* __restrict__ bb, const float* __restrict__ bnstats,
    float invN, float* __restrict__ Z, int zoff) {
  __shared__ float rs[256], rq[256];
  const int node = blockIdx.x * 4 + (threadIdx.x >> 6);
  const int f = threadIdx.x & 63;
  const float h = H[(size_t)node * 64 + f];
  float mu, var;
  if (bnstats) {
    mu = bnstats[f] * invN;
    var = bnstats[64 + f] * invN - mu * mu;
  } else {
    rs[threadIdx.x] = h;
    rq[threadIdx.x] = h * h;
    __syncthreads();
#pragma unroll
    for (int off = 32; off > 0; off >>= 1) {
      if (f < off) {
        rs[threadIdx.x] += rs[threadIdx.x + off];
        rq[threadIdx.x] += rq[threadIdx.x + off];
      }
      __syncthreads();
    }
    const int base = threadIdx.x & ~63;
    mu = rs[base] * (1.f / 64.f);
    var = rq[base] * (1.f / 64.f) - mu * mu;
  }
  const float v = (h - mu) * rsqrtf(var + 1e-5f) * gg[f] + bb[f];
  Z[(size_t)node * 192 + zoff + f] = fmaxf(v, 0.f);
}

// -------------------- inner head: colsum(relu(C@W + b)) --------------------
__global__ __launch_bounds__(256) void eric_inner_sum_kernel(
    const float* __restrict__ X, int xstride, const float* __restrict__ W,
    const float* __restrict__ B, float* __restrict__ dsum) {
  __shared__ float sW[64 * 64];
  __shared__ float sB[64];
  __shared__ float sX[8][16 * XPAD];
  const int tid = threadIdx.x, wid = tid >> 5, lane = tid & 31;
  const int half = lane >> 4, l15 = lane & 15;
  for (int i = tid; i < 4096; i += 256) sW[i] = W[i];
  if (tid < 64) sB[tid] = B[tid];
  __syncthreads();
  const int tile = blockIdx.x * 8 + wid;
  if (tile >= TILES_M) return;          // wave-uniform
  const int node0 = tile * 16;
  float* sx = &sX[wid][0];
  for (int i = lane; i < 256; i += 32) {
    int r = i >> 4, c4 = i & 15;
    const float4 xv = *reinterpret_cast<const float4*>(
        X + (size_t)(node0 + r) * xstride + c4 * 4);
    sx[r * XPAD + c4 * 4 + 0] = xv.x;
    sx[r * XPAD + c4 * 4 + 1] = xv.y;
    sx[r * XPAD + c4 * 4 + 2] = xv.z;
    sx[r * XPAD + c4 * 4 + 3] = xv.w;
  }
  v8f acc[4] = {{}, {}, {}, {}};
#pragma unroll
  for (int kk = 0; kk < 16; ++kk) {
    const int kb = kk * 4 + half * 2;
    v2f a; a.x = sx[l15 * XPAD + kb]; a.y = sx[l15 * XPAD + kb + 1];
#pragma unroll
    for (int nt = 0; nt < 4; ++nt) {
      const int col = nt * 16 + l15;
      v2f b; b.x = sW[kb * 64 + col]; b.y = sW[(kb + 1) * 64 + col];
      acc[nt] = wmma_f32_16x16x4(a, b, acc[nt]);
    }
  }
#pragma unroll
  for (int nt = 0; nt < 4; ++nt) {
    const int col = nt * 16 + l15;
    float p = 0.f;
#pragma unroll
    for (int i = 0; i < 8; ++i) p += fmaxf(acc[nt][i] + sB[col], 0.f);
    p += __shfl_down(p, 16, 32);        // lanes l and l+16 own the same col
    if (half == 0) atomicAdd(&dsum[col], p);
  }
}

// ----------------- outer head + per-layer diff (tiny, 64 thr) --------------
__global__ void eric_outer_kernel(const float* __restrict__ d1,
                                  const float* __restrict__ d2,
                                  const float* __restrict__ Wo,
                                  const float* __restrict__ bo,
                                  float* __restrict__ o1, float* __restrict__ o2,
                                  float* __restrict__ diff) {
  const int f = threadIdx.x;  // 64
  float a = bo[f], b = bo[f];
  for (int k = 0; k < 64; ++k) {
    const float w = Wo[k * 64 + f];
    a += d1[k] * w;
    b += d2[k] * w;
  }
  a = fmaxf(a, 0.f);
  b = fmaxf(b, 0.f);
  o1[f] = a;
  o2[f] = b;
  const float d = a - b;
  diff[f] = expf(-d * d);
}

// ------------------------ graph embedding column sums ----------------------
__global__ void eric_colsum_kernel(const float* __restrict__ Z,
                                   float* __restrict__ gvec) {
  const int t = threadIdx.x;  // 192
  const int n0 = blockIdx.x * 500;
  float acc = 0.f;
  for (int n = n0; n < n0 + 500; ++n) acc += Z[(size_t)n * 192 + t];
  atomicAdd(&gvec[t], acc);
}

// ---------------------------- contrastive terms ----------------------------
__device__ __forceinline__ float eric_pe(float s) {
  // log(2) - softplus(-s), numerically stable softplus
  const float x = -s;
  const float sp = fmaxf(x, 0.f) + log1pf(expf(-fabsf(x)));
  return 0.6931471805599453f - sp;
}

__global__ __launch_bounds__(256) void eric_cl_kernel(
    const float* __restrict__ Z, const float* __restrict__ ga,
    const float* __restrict__ gb, float* __restrict__ accA,
    float* __restrict__ accB) {
  __shared__ float sA, sB;
  if (threadIdx.x == 0) { sA = 0.f; sB = 0.f; }
  __syncthreads();
  const int wid = threadIdx.x >> 5, lane = threadIdx.x & 31;
  const int node = blockIdx.x * 8 + wid;
  float d1 = 0.f, d2 = 0.f;
#pragma unroll
  for (int j = 0; j < 6; ++j) {
    const int c = lane + 32 * j;
    const float z = Z[(size_t)node * 192 + c];
    d1 += z * ga[c];
    d2 += z * gb[c];
  }
#pragma unroll
  for (int off = 16; off > 0; off >>= 1) {
    d1 += __shfl_down(d1, off, 32);
    d2 += __shfl_down(d2, off, 32);
  }
  if (lane == 0) {
    atomicAdd(&sA, eric_pe(d1));
    atomicAdd(&sB, eric_pe(d2));
  }
  __syncthreads();
  if (threadIdx.x == 0) {
    atomicAdd(accA, sA);
    atomicAdd(accB, sB);
  }
}

// ------------------------------- final head --------------------------------
__global__ __launch_bounds__(256) void eric_head_kernel(
    const float* __restrict__ diff, const float* __restrict__ o1,
    const float* __restrict__ o2, const float* __restrict__ S,
    const float* __restrict__ cs1w, const float* __restrict__ cs1b,
    const float* __restrict__ cs2w, const float* __restrict__ cs2b,
    const float* __restrict__ sl1w, const float* __restrict__ sl1b,
    const float* __restrict__ sl2w, const float* __restrict__ sl2b,
    const float* __restrict__ W3, const float* __restrict__ Vm,
    const float* __restrict__ b3, const float* __restrict__ ss1w,
    const float* __restrict__ ss1b, const float* __restrict__ ss2w,
    const float* __restrict__ ss2b, const float* __restrict__ alpha,
    const float* __restrict__ beta, const float* __restrict__ gamma,
    float* __restrict__ out) {
  __shared__ float sr[96], srep[96], s16[16], t16[16], simr[16], red[256];
  const int t = threadIdx.x;
  if (t < 96) {  // conv_stack layer 1
    float a = cs1b[t];
    for (int k = 0; k < 192; ++k) a += diff[k] * cs1w[k * 96 + t];
    sr[t] = fmaxf(a, 0.f);
  }
  __syncthreads();
  if (t < 96) {  // conv_stack layer 2 (tanh)
    float a = cs2b[t];
    for (int k = 0; k < 96; ++k) a += sr[k] * cs2w[k * 96 + t];
    srep[t] = tanhf(a);
  }
  __syncthreads();
  if (t < 16) {  // score head hidden
    float a = sl1b[t];
    for (int k = 0; k < 96; ++k) a += srep[k] * sl1w[k * 16 + t];
    s16[t] = fmaxf(a, 0.f);
  }
  {  // NTN term1[k] = o1 . W3[k] . o2 — 16 k-slices x 16 row-chunks
    const int k = t >> 4, c = t & 15;
    float a = 0.f;
    for (int d = c * 4; d < c * 4 + 4; ++d) {
      const float* w = &W3[k * 4096 + d * 64];
      float s = 0.f;
      for (int e = 0; e < 64; ++e) s += w[e] * o2[e];
      a += o1[d] * s;
    }
    red[t] = a;
  }
  __syncthreads();
  if (t < 16) {
    float a = 0.f;
    for (int c = 0; c < 16; ++c) a += red[t * 16 + c];
    float t2 = 0.f;
    for (int j = 0; j < 64; ++j) {
      t2 += o1[j] * Vm[t * 128 + j];
      t2 += o2[j] * Vm[t * 128 + 64 + j];
    }
    simr[t] = fmaxf(a + t2 + b3[t], 0.f);
  }
  __syncthreads();
  if (t < 16) {
    float a = ss1b[t];
    for (int k = 0; k < 16; ++k) a += simr[k] * ss1w[k * 16 + t];
    t16[t] = fmaxf(a, 0.f);
  }
  __syncthreads();
  if (t == 0) {
    float ss = ss2b[0];
    for (int k = 0; k < 16; ++k) ss += t16[k] * ss2w[k];
    const float sim = 1.f / (1.f + expf(-ss));
    float sc = sl2b[0];
    for (int k = 0; k < 16; ++k) sc += s16[k] * sl2w[k];
    const float score = 1.f / (1.f + expf(-sc));
    out[0] = alpha[0] * score + beta[0] * sim;
    const float Ldiff = (S[0] - S[1]) - (S[2] - S[3]);  // L1 - L2
    out[1] = tanhf(Ldiff * gamma[0]);
  }
}

// ---------------------------------------------------------------------------

extern "C" void kernel_launch(void* const* d_in, const int* in_sizes, int n_in,
                              void* d_out, int out_size, void* d_ws,
                              size_t ws_size, hipStream_t stream) {
  // setup_inputs order: x1, x2, params (pytree leaves, dict keys sorted), e1, e2
  const float* x[2] = {(const float*)d_in[0], (const float*)d_in[1]};
  const float* Vm    = (const float*)d_in[2];   // V [16,128]
  const float* W3    = (const float*)d_in[3];   // [16,64,64]
  const float* alpha = (const float*)d_in[4];
  const float* b3    = (const float*)d_in[5];   // [1,16]
  const float* beta  = (const float*)d_in[6];
  const float* cs1b  = (const float*)d_in[7];
  const float* cs1w  = (const float*)d_in[8];   // [192,96]
  const float* cs2b  = (const float*)d_in[9];
  const float* cs2w  = (const float*)d_in[10];  // [96,96]
  const float* gamma = (const float*)d_in[11];
  // gin[i]: b1,b2,bt,g,w1,w2 at 12+6i ; inner[i]: b,w at 30+2i ; outer: 36+2i
  const float* sl1b = (const float*)d_in[42];
  const float* sl1w = (const float*)d_in[43];   // [96,16]
  const float* sl2b = (const float*)d_in[44];
  const float* sl2w = (const float*)d_in[45];   // [16,1]
  const float* ss1b = (const float*)d_in[46];
  const float* ss1w = (const float*)d_in[47];   // [16,16]
  const float* ss2b = (const float*)d_in[48];
  const float* ss2w = (const float*)d_in[49];   // [16,1]
  const int* E[2] = {(const int*)d_in[50], (const int*)d_in[51]};

  // workspace layout (floats): Z1 | Z2 | AGG | HBUF | small
  float* ws = (float*)d_ws;
  float* Z[2] = {ws, ws + (size_t)N_NODES * 192};
  float* AGG  = ws + 2 * (size_t)N_NODES * 192;
  float* HB   = AGG + (size_t)N_NODES * 64;
  float* SM   = HB + (size_t)N_NODES * 64;
  float* bnstats = SM;                 // [128] sum|sumsq
  float* dsum[2] = {SM + 128, SM + 192};
  float* o1 = SM + 256;
  float* o2 = SM + 320;
  float* diff = SM + 384;              // [192]
  float* g1 = SM + 576;                // [192]
  float* g2 = SM + 768;                // [192]
  float* Sacc = SM + 960;              // S11,S12,S22,S21

  eric_zero_kernel<<<4, 256, 0, stream>>>(SM, 1024);

  for (int i = 0; i < 3; ++i) {
    const float* b1 = (const float*)d_in[12 + 6 * i + 0];
    const float* b2 = (const float*)d_in[12 + 6 * i + 1];
    const float* bt = (const float*)d_in[12 + 6 * i + 2];
    const float* gg = (const float*)d_in[12 + 6 * i + 3];
    const float* w1 = (const float*)d_in[12 + 6 * i + 4];
    const float* w2 = (const float*)d_in[12 + 6 * i + 5];
    const float* ib = (const float*)d_in[30 + 2 * i + 0];
    const float* iw = (const float*)d_in[30 + 2 * i + 1];
    const float* ob = (const float*)d_in[36 + 2 * i + 0];
    const float* ow = (const float*)d_in[36 + 2 * i + 1];

    for (int gidx = 0; gidx < 2; ++gidx) {
      const float* Xin;
      int xs;
      if (i == 0) { Xin = x[gidx]; xs = 64; }
      else        { Xin = Z[gidx] + 64 * (i - 1); xs = 192; }

      eric_zero_kernel<<<12500, 256, 0, stream>>>(AGG, N_NODES * 64);
      if (i == 0) eric_zero_kernel<<<1, 256, 0, stream>>>(bnstats, 128);
      eric_scatter_kernel<<<100000, 256, 0, stream>>>(Xin, xs, E[gidx], AGG);
      eric_gin_gemm_kernel<<<GEMM_BLOCKS, 256, 0, stream>>>(
          Xin, xs, AGG, w1, b1, w2, b2, HB, i == 0 ? bnstats : nullptr);
      eric_norm_kernel<<<12500, 256, 0, stream>>>(
          HB, gg, bt, i == 0 ? bnstats : nullptr, 1.f / N_NODES, Z[gidx], 64 * i);
      eric_zero_kernel<<<1, 256, 0, stream>>>(dsum[gidx], 64);
      eric_inner_sum_kernel<<<GEMM_BLOCKS, 256, 0, stream>>>(
          Z[gidx] + 64 * i, 192, iw, ib, dsum[gidx]);
    }
    eric_outer_kernel<<<1, 64, 0, stream>>>(dsum[0], dsum[1], ow, ob, o1, o2,
                                            diff + 64 * i);
  }

  // g1 = z1.sum(0), g2 = z2.sum(0)
  eric_colsum_kernel<<<100, 192, 0, stream>>>(Z[0], g1);
  eric_colsum_kernel<<<100, 192, 0, stream>>>(Z[1], g2);
  // S11=sum pe(z1.g1), S12=sum pe(z1.g2), S22=sum pe(z2.g2), S21=sum pe(z2.g1)
  eric_cl_kernel<<<6250, 256, 0, stream>>>(Z[0], g1, g2, &Sacc[0], &Sacc[1]);
  eric_cl_kernel<<<6250, 256, 0, stream>>>(Z[1], g2, g1, &Sacc[2], &Sacc[3]);

  eric_head_kernel<<<1, 256, 0, stream>>>(
      diff, o1, o2, Sacc, cs1w, cs1b, cs2w, cs2b, sl1w, sl1b, sl2w, sl2b, W3,
      Vm, b3, ss1w, ss1b, ss2w, ss2b, alpha, beta, gamma, (float*)d_out);
}